// SConv2d_v1_13065290515015
// MI455X (gfx1250) — compile-verified
//
#include <hip/hip_runtime.h>
#include <stdint.h>

// SConv2d_v1: unfold(3x3, pad1) -> elementwise * weight -> MAJ3 tree (27->9->3->1)
// MAJ3(a,b,c) == (a + b + c - a*b*c) / 2   (exact closed form of the subset sum)
//
// Shapes: x[2,3,224,224] f32, w[32,3,3,3] f32, out[2,32,224,224] f32.
// Store-dominated (12.8 MB out vs 1.2 MB in); tile input through LDS via the
// gfx1250 async global->LDS path, keep the 27-value patch in VGPRs, reuse
// across all 32 output channels.

#define C_IN    3
#define H_IN    224
#define W_IN    224
#define O_OUT   32
#define N_BATCH 2

#define TW   32
#define TH   8
#define TW2  (TW + 2)                 // 34
#define TH2  (TH + 2)                 // 10
#define CH_ELEMS   (TW2 * TH2)        // 340
#define TILE_ELEMS (C_IN * CH_ELEMS)  // 1020
#define W_ELEMS    (O_OUT * C_IN * 9) // 864
#define NTHREADS   256

#if defined(__has_builtin)
#if __has_builtin(__builtin_amdgcn_global_load_async_to_lds_b32)
#define HAVE_ASYNC_LDS 1
#endif
#if __has_builtin(__builtin_amdgcn_s_wait_asynccnt)
#define HAVE_WAIT_ASYNC 1
#endif
#endif

// Per hipcc diagnostic, the builtin's first param is a (non-const) addrspace(1)
// int*; the LDS side is addrspace(3) int*.
typedef __attribute__((address_space(1))) int GI;
typedef __attribute__((address_space(3))) int LI;

// Async copy one dword global -> LDS (falls back to load+ds_store if the
// builtin is unavailable on this toolchain).
__device__ __forceinline__ void async_copy_b32(const float* __restrict__ g,
                                               float* __restrict__ l) {
#if defined(HAVE_ASYNC_LDS)
  __builtin_amdgcn_global_load_async_to_lds_b32(
      (GI*)(uintptr_t)g,                 // global addr (AS1) == generic bits
      (LI*)(uint32_t)(uintptr_t)l,       // LDS byte offset == addr[31:0]
      0, 0);
#else
  *l = *g;
#endif
}

__device__ __forceinline__ void wait_async_zero() {
#if defined(HAVE_ASYNC_LDS)
#if defined(HAVE_WAIT_ASYNC)
  __builtin_amdgcn_s_wait_asynccnt(0);
#else
  asm volatile("s_wait_asynccnt 0x0" ::: "memory");
#endif
#endif
}

__device__ __forceinline__ float maj3(float a, float b, float c) {
  return 0.5f * ((a + b + c) - a * b * c);
}

__global__ __launch_bounds__(NTHREADS) void sconv_maj_kernel(
    const float* __restrict__ x, const float* __restrict__ wgt,
    float* __restrict__ out) {
  __shared__ float tile[TILE_ELEMS];  // [c][y][x] halo tile, pitch 34
  __shared__ float wl[W_ELEMS];       // weights, o-major (27 per o)

  const int tid = threadIdx.x;
  const int w0  = blockIdx.x * TW;    // 0..6  * 32
  const int h0  = blockIdx.y * TH;    // 0..27 * 8
  const int n   = blockIdx.z;         // 0..1
  const float* xn = x + (size_t)n * (C_IN * H_IN * W_IN);

  // ---- stage weights (all in-bounds) ----
  for (int e = tid; e < W_ELEMS; e += NTHREADS)
    async_copy_b32(wgt + e, wl + e);

  // ---- stage input halo tile; zero-pad out-of-image lanes ----
  for (int e = tid; e < TILE_ELEMS; e += NTHREADS) {
    const int c  = e / CH_ELEMS;
    const int r  = e - c * CH_ELEMS;
    const int y  = r / TW2;
    const int xx = r - y * TW2;
    const int gh = h0 + y - 1;
    const int gw = w0 + xx - 1;
    const bool inb = (gh >= 0) & (gh < H_IN) & (gw >= 0) & (gw < W_IN);
    if (inb) {
      async_copy_b32(xn + ((size_t)c * H_IN + gh) * W_IN + gw, tile + e);
    } else {
      tile[e] = 0.0f;
    }
  }
  wait_async_zero();   // drain this wave's ASYNCcnt before the barrier
  __syncthreads();     // (compiler also drains DScnt here)

  // ---- gather this thread's 3x3x3 patch into registers ----
  const int tx = tid & (TW - 1);
  const int ty = tid >> 5;
  float p[3][3][3];
#pragma unroll
  for (int c = 0; c < 3; ++c)
#pragma unroll
    for (int ki = 0; ki < 3; ++ki)
#pragma unroll
      for (int kj = 0; kj < 3; ++kj)
        p[c][ki][kj] = tile[c * CH_ELEMS + (ty + ki) * TW2 + (tx + kj)];

  const int hh = h0 + ty;
  const int ww = w0 + tx;
  float* outp = out + ((size_t)n * O_OUT) * (size_t)(H_IN * W_IN)
                    + (size_t)hh * W_IN + ww;

  // ---- 32 output channels from registers; coalesced 128B stores per wave ----
#pragma unroll 4
  for (int o = 0; o < O_OUT; ++o) {
    const float* wo = wl + o * 27;
    float l2[3];
#pragma unroll
    for (int c = 0; c < 3; ++c) {
      float l1[3];
#pragma unroll
      for (int ki = 0; ki < 3; ++ki) {
        const float a = p[c][ki][0] * wo[c * 9 + ki * 3 + 0];
        const float b = p[c][ki][1] * wo[c * 9 + ki * 3 + 1];
        const float d = p[c][ki][2] * wo[c * 9 + ki * 3 + 2];
        l1[ki] = maj3(a, b, d);
      }
      l2[c] = maj3(l1[0], l1[1], l1[2]);
    }
    outp[(size_t)o * (size_t)(H_IN * W_IN)] = maj3(l2[0], l2[1], l2[2]);
  }
}

extern "C" void kernel_launch(void* const* d_in, const int* in_sizes, int n_in,
                              void* d_out, int out_size, void* d_ws, size_t ws_size,
                              hipStream_t stream) {
  (void)in_sizes; (void)n_in; (void)out_size; (void)d_ws; (void)ws_size;
  const float* x   = (const float*)d_in[0];
  const float* wgt = (const float*)d_in[1];
  float* out       = (float*)d_out;

  dim3 grid(W_IN / TW, H_IN / TH, N_BATCH);  // (7, 28, 2)
  sconv_maj_kernel<<<grid, NTHREADS, 0, stream>>>(x, wgt, out);
}